// Relation_25031069401586
// MI455X (gfx1250) — compile-verified
//
#include <hip/hip_runtime.h>
#include <hip/hip_bf16.h>
#include <math.h>

// Sizes fixed by the reference
#define NN    2048      // N rows
#define MM    2048      // nongt rows
#define EMBD  64
#define GRP   16
#define FEATD 256
#define DMD   1024
#define DH    64        // DM / GROUP

typedef __attribute__((ext_vector_type(2))) float v2f;
typedef __attribute__((ext_vector_type(8))) float v8f;

static __device__ __forceinline__ v8f wmma4(v2f a, v2f b, v8f c) {
  // D = A(16x4) * B(4x16) + C(16x16), fp32 WMMA
  return __builtin_amdgcn_wmma_f32_16x16x4_f32(false, a, false, b, (short)0, c,
                                               false, false);
}

// ---------------------------------------------------------------------------
// Generic f32 WMMA GEMM: C[m][n] = sum_k A[m][k] * B(k,n) (+ bias[n])
//   BT=true : B stored row-major [N][K]  (computes A*B^T)
//   BT=false: B stored row-major [K][N]  (computes A*B)
// One wave per block; each wave produces a 16 x (16*NT) tile of C.
// grid.z batches with per-z element offsets aZ (on A) and cZ (on C).
// ---------------------------------------------------------------------------
template <bool BT, int NT>
__global__ __launch_bounds__(32) void wmma_gemm_f32(
    const float* __restrict__ A, const float* __restrict__ B,
    const float* __restrict__ bias, float* __restrict__ C,
    int K, int lda, int ldb, int ldc, long aZ, long cZ) {
  const int lane = threadIdx.x & 31;
  const int half = lane >> 4;      // 0: lanes 0-15, 1: lanes 16-31
  const int l16  = lane & 15;
  const int m0 = blockIdx.y * 16;
  const int n0 = blockIdx.x * (16 * NT);

  A += (size_t)blockIdx.z * (size_t)aZ;
  C += (size_t)blockIdx.z * (size_t)cZ;

  const float* __restrict__ arow = A + (size_t)(m0 + l16) * lda;

  v8f acc[NT];
#pragma unroll
  for (int t = 0; t < NT; ++t) acc[t] = {};

  for (int kk = 0; kk < K; kk += 4) {
    const int kidx = kk + 2 * half;
    v2f a;
    a.x = arow[kidx];
    a.y = arow[kidx + 1];
#pragma unroll
    for (int t = 0; t < NT; ++t) {
      const int n = n0 + t * 16 + l16;
      v2f b;
      if (BT) {
        b.x = B[(size_t)n * ldb + kidx];
        b.y = B[(size_t)n * ldb + kidx + 1];
      } else {
        b.x = B[(size_t)kidx * ldb + n];
        b.y = B[(size_t)(kidx + 1) * ldb + n];
      }
      acc[t] = wmma4(a, b, acc[t]);
    }
  }

#pragma unroll
  for (int t = 0; t < NT; ++t) {
    const int n = n0 + t * 16 + l16;
    const float bv = bias ? bias[n] : 0.0f;
#pragma unroll
    for (int r = 0; r < 8; ++r) {
      const int m = m0 + r + 8 * half;
      C[(size_t)m * ldc + n] = acc[t][r] + bv;  // coalesced over lanes (n)
    }
  }
}

// ---------------------------------------------------------------------------
// Fused attention-logit kernel.
// One wave per (16n x 16m) tile. Produces
//   weighted[n][g][m] = log(max(PE[n][m][:]·fc1w[g][:] + fc1b[g], 1e-6))
//                     + (1/sqrt(64)) * sum_d q[n][g*64+d] * k[m][g*64+d]
// Phase A: 16 WMMA tiles (m x g) of the position term -> LDS (padded).
// Phase B: 16 per-group WMMA tiles (n x m) of q·k^T, add LDS term, store.
// PE is streamed exactly once from HBM.
// ---------------------------------------------------------------------------
__global__ __launch_bounds__(32) void fused_logits(
    const float* __restrict__ pe,    // [NN][MM][EMBD]
    const float* __restrict__ fc1w,  // [GRP][EMBD]
    const float* __restrict__ fc1b,  // [GRP]
    const float* __restrict__ q,     // [NN][DMD]
    const float* __restrict__ k,     // [MM][DMD]
    float* __restrict__ wgt)         // [NN][GRP][MM]
{
  __shared__ float lw[16][16][17];  // [n_local][m_local][g], +1 pad vs banks

  const int lane = threadIdx.x & 31;
  const int half = lane >> 4;
  const int l16  = lane & 15;
  const int n0 = blockIdx.y * 16;
  const int m0 = blockIdx.x * 16;

  // ---- Phase A: position-bias tiles, one per local n row -----------------
  const float bg = fc1b[l16];  // lane's group bias (N-dim of this WMMA = g)
#pragma unroll 1
  for (int nl = 0; nl < 16; ++nl) {
    const float* __restrict__ perow =
        pe + ((size_t)(n0 + nl) * MM + m0) * EMBD;
    v8f acc = {};
#pragma unroll
    for (int kk = 0; kk < EMBD; kk += 4) {
      const int kidx = kk + 2 * half;
      v2f a, b;
      a.x = perow[(size_t)l16 * EMBD + kidx];      // A[m=l16][e]
      a.y = perow[(size_t)l16 * EMBD + kidx + 1];
      b.x = fc1w[l16 * EMBD + kidx];               // B[e][g=l16] = fc1w[g][e]
      b.y = fc1w[l16 * EMBD + kidx + 1];
      acc = wmma4(a, b, acc);
    }
    // acc[r] = pos_lin[m = r+8*half][g = l16]
#pragma unroll
    for (int r = 0; r < 8; ++r) {
      const float z = acc[r] + bg;
      lw[nl][r + 8 * half][l16] = __logf(fmaxf(z, 1e-6f));
    }
  }
  __syncthreads();

  // ---- Phase B: q·k^T per group, combine, store coalesced ----------------
  const float scale = 0.125f;  // 1/sqrt(64)
  const float* __restrict__ qrow = q + (size_t)(n0 + l16) * DMD;
  const float* __restrict__ krow = k + (size_t)(m0 + l16) * DMD;
#pragma unroll 1
  for (int g = 0; g < GRP; ++g) {
    v8f acc = {};
#pragma unroll
    for (int kk = 0; kk < DH; kk += 4) {
      const int kidx = g * DH + kk + 2 * half;
      v2f a, b;
      a.x = qrow[kidx];      // A[n=l16][d]
      a.y = qrow[kidx + 1];
      b.x = krow[kidx];      // B[d][m=l16] = k[m][g*64+d]
      b.y = krow[kidx + 1];
      acc = wmma4(a, b, acc);
    }
    // acc[r] = aff[n = r+8*half][m = l16]
#pragma unroll
    for (int r = 0; r < 8; ++r) {
      const int nl = r + 8 * half;
      const float v = acc[r] * scale + lw[nl][l16][g];
      wgt[((size_t)(n0 + nl) * GRP + g) * MM + m0 + l16] = v;
    }
  }
}

// ---------------------------------------------------------------------------
// In-place row softmax, row length 2048, one wave per row.
// ---------------------------------------------------------------------------
__global__ __launch_bounds__(32) void softmax2048(float* __restrict__ w) {
  float* __restrict__ p = w + (size_t)blockIdx.x * MM;
  const int lane = threadIdx.x & 31;

  float v[64];
  float mx = -3.402823466e38f;
#pragma unroll
  for (int i = 0; i < 64; ++i) {
    v[i] = p[lane + 32 * i];
    mx = fmaxf(mx, v[i]);
  }
#pragma unroll
  for (int off = 16; off >= 1; off >>= 1)
    mx = fmaxf(mx, __shfl_xor(mx, off, 32));

  float s = 0.0f;
#pragma unroll
  for (int i = 0; i < 64; ++i) {
    v[i] = __expf(v[i] - mx);
    s += v[i];
  }
#pragma unroll
  for (int off = 16; off >= 1; off >>= 1) s += __shfl_xor(s, off, 32);

  const float inv = 1.0f / s;
#pragma unroll
  for (int i = 0; i < 64; ++i) p[lane + 32 * i] = v[i] * inv;
}

// ---------------------------------------------------------------------------
extern "C" void kernel_launch(void* const* d_in, const int* in_sizes, int n_in,
                              void* d_out, int out_size, void* d_ws,
                              size_t ws_size, hipStream_t stream) {
  const float* roi   = (const float*)d_in[0];  // [NN][FEATD]
  const float* pe    = (const float*)d_in[1];  // [NN][MM][EMBD]
  const float* fc1w  = (const float*)d_in[2];  // [GRP][EMBD]
  const float* fc1b  = (const float*)d_in[3];  // [GRP]
  const float* fc2w  = (const float*)d_in[4];  // [DMD][FEATD]
  const float* fc2b  = (const float*)d_in[5];  // [DMD]
  const float* fc3w  = (const float*)d_in[6];  // [DMD][FEATD]
  const float* fc3b  = (const float*)d_in[7];  // [DMD]
  const float* convw = (const float*)d_in[8];  // [FEATD][GRP*FEATD]
  const float* convb = (const float*)d_in[9];  // [FEATD]
  float* out = (float*)d_out;                  // [NN][FEATD]

  // Workspace layout (floats): q | k | logits | out_t   (~304 MB total)
  float* qbuf = (float*)d_ws;
  float* kbuf = qbuf + (size_t)NN * DMD;
  float* wsm  = kbuf + (size_t)MM * DMD;
  float* outt = wsm + (size_t)NN * GRP * MM;

  // K1: q = roi * fc2w^T + fc2b        (2048x256)x(256x1024)
  wmma_gemm_f32<true, 2><<<dim3(DMD / 32, NN / 16, 1), 32, 0, stream>>>(
      roi, fc2w, fc2b, qbuf, FEATD, FEATD, FEATD, DMD, 0, 0);

  // K2: k = roi * fc3w^T + fc3b
  wmma_gemm_f32<true, 2><<<dim3(DMD / 32, MM / 16, 1), 32, 0, stream>>>(
      roi, fc3w, fc3b, kbuf, FEATD, FEATD, FEATD, DMD, 0, 0);

  // K3: fused position-bias + q·k^T logits  -> wsm[n][g][m]
  fused_logits<<<dim3(MM / 16, NN / 16, 1), 32, 0, stream>>>(
      pe, fc1w, fc1b, qbuf, kbuf, wsm);

  // K4: softmax over m for each of NN*GRP rows
  softmax2048<<<dim3(NN * GRP, 1, 1), 32, 0, stream>>>(wsm);

  // K5: out_t[n][g][c] = sum_m softmax[n][g][m] * roi[m][c]  (per-g, grid.z)
  wmma_gemm_f32<false, 4><<<dim3(FEATD / 64, NN / 16, GRP), 32, 0, stream>>>(
      wsm, roi, nullptr, outt,
      /*K=*/MM, /*lda=*/GRP * MM, /*ldb=*/FEATD, /*ldc=*/GRP * FEATD,
      /*aZ=*/(long)MM, /*cZ=*/(long)FEATD);

  // K6: out = out_t_flat * convw^T + convb   (2048x4096)x(4096x256)
  wmma_gemm_f32<true, 2><<<dim3(FEATD / 32, NN / 16, 1), 32, 0, stream>>>(
      outt, convw, convb, out, GRP * FEATD, GRP * FEATD, GRP * FEATD, FEATD, 0,
      0);
}